// GAT_dif_2491081032177
// MI455X (gfx1250) — compile-verified
//
#include <hip/hip_runtime.h>
#include <cstdint>
#include <cstddef>

// ---------------------------------------------------------------------------
// GAT forward (2 branches, 6 GAT conv layers) for MI455X (gfx1250, wave32).
// GEMMs use V_WMMA_F32_16X16X4_F32 (fp32 matrix pipe -> reference precision).
// Each wave computes a 16 x (16*NT) strip: 1 A-frag load feeds NT WMMAs.
// Edge softmax + aggregation use L2-resident float atomics (feat/agg tables
// are ~100MB << 192MB L2); the gather side uses b128 vector loads.
// ---------------------------------------------------------------------------

typedef __attribute__((ext_vector_type(2))) float v2f;
typedef __attribute__((ext_vector_type(8))) float v8f;

enum : int { NN = 50000, NE = 800000, FIN = 256, H1 = 64, NC = 40, NCP = 48 };

static inline int cdiv(long long a, long long b) { return (int)((a + b - 1) / b); }

// ---- monotone float <-> uint mapping for atomic float max -----------------
__device__ __forceinline__ unsigned fmap(float f) {
    unsigned u = __float_as_uint(f);
    return (u & 0x80000000u) ? ~u : (u | 0x80000000u);
}
__device__ __forceinline__ float funmap(unsigned u) {
    unsigned b = (u & 0x80000000u) ? (u & 0x7FFFFFFFu) : ~u;
    return __uint_as_float(b);
}

// ---------------------------------------------------------------------------
// fp32 WMMA GEMM: C[M x FoutP] = A[M x K] * B[K x FoutP]
// One wave per 16 x (16*NT) strip. grid = (M/16, FoutP/(16*NT)), block = 32.
// A-layout (16x4 f32): lanes 0-15 -> K {k,k+1}; lanes 16-31 -> K {k+2,k+3}.
// B-layout (4x16 f32): vgpr0/1 hold rows (k+2*half)/(k+2*half+1), N striped.
// C-layout: VGPR v -> row (v + 8*half), col = lane&15.
// ---------------------------------------------------------------------------
template <int NT>
__global__ void wmma_gemm_f32(const float* __restrict__ A, int lda,
                              const float* __restrict__ B, int ldb,
                              float* __restrict__ C, int ldc, int K) {
    const int lane = threadIdx.x;      // 0..31, wave32
    const int half = lane >> 4;        // 0 or 1
    const int l16  = lane & 15;
    const int m0 = blockIdx.x * 16;
    const int n0 = blockIdx.y * 16 * NT;

    v8f acc[NT];
#pragma unroll
    for (int t = 0; t < NT; ++t) acc[t] = (v8f){};

    const float* arow  = A + (size_t)(m0 + l16) * lda + 2 * half;
    const float* bbase = B + n0 + l16;

    for (int k = 0; k < K; k += 4) {
        v2f a;
        a.x = arow[k];
        a.y = arow[k + 1];
        const float* brow0 = bbase + (size_t)(k + 2 * half)     * ldb;
        const float* brow1 = bbase + (size_t)(k + 2 * half + 1) * ldb;
#pragma unroll
        for (int t = 0; t < NT; ++t) {
            v2f b;
            b.x = brow0[t * 16];
            b.y = brow1[t * 16];
            acc[t] = __builtin_amdgcn_wmma_f32_16x16x4_f32(
                /*neg_a=*/false, a, /*neg_b=*/false, b,
                /*c_mod=*/(short)0, acc[t], /*reuse_a=*/false, /*reuse_b=*/false);
        }
    }

#pragma unroll
    for (int t = 0; t < NT; ++t) {
        float* crow = C + (size_t)(m0 + 8 * half) * ldc + n0 + t * 16 + l16;
#pragma unroll
        for (int v = 0; v < 8; ++v) crow[(size_t)v * ldc] = acc[t][v];
    }
}

// ---- zero-pad weight matrix [K x Fout] -> [K x FoutP] ---------------------
__global__ void pad_w_kernel(const float* __restrict__ W, float* __restrict__ Wp,
                             int K, int Fout, int FoutP) {
    int idx = blockIdx.x * blockDim.x + threadIdx.x;
    if (idx >= K * FoutP) return;
    int r = idx / FoutP, c = idx % FoutP;
    Wp[idx] = (c < Fout) ? W[(size_t)r * Fout + c] : 0.0f;
}

// ---- per-node attention logits: el/er[n,h] = <feat[n,h,:], al/ar[h,:]> ----
__global__ void el_er_kernel(const float* __restrict__ feat, int ldf, int H, int D,
                             const float* __restrict__ al, const float* __restrict__ ar,
                             float* __restrict__ el, float* __restrict__ er, int N) {
    int idx = blockIdx.x * blockDim.x + threadIdx.x;
    if (idx >= N * H) return;
    int n = idx / H, h = idx % H;
    const float* f = feat + (size_t)n * ldf + h * D;
    float sl = 0.f, sr = 0.f;
    for (int d = 0; d < D; ++d) {
        float v = f[d];
        sl += v * al[h * D + d];
        sr += v * ar[h * D + d];
    }
    el[idx] = sl;
    er[idx] = sr;
}

__global__ void init_seg_kernel(unsigned* __restrict__ emaxu, float* __restrict__ esum, int NH) {
    int i = blockIdx.x * blockDim.x + threadIdx.x;
    if (i < NH) { emaxu[i] = 0x007FFFFFu /* fmap(-inf) */; esum[i] = 0.f; }
}

__global__ void zero_f_kernel(float* __restrict__ p, long long n) {
    long long i = (long long)blockIdx.x * blockDim.x + threadIdx.x;
    if (i < n) p[i] = 0.f;
}

// ---- edge pass 1: leaky_relu(el[src]+er[dst]); segment max via atomic -----
__global__ void edge_pass1(const int* __restrict__ src, const int* __restrict__ dst,
                           const float* __restrict__ el, const float* __restrict__ er,
                           float* __restrict__ eval, unsigned* __restrict__ emaxu, int H) {
    long long idx = (long long)blockIdx.x * blockDim.x + threadIdx.x;
    if (idx >= (long long)NE * H) return;
    int e = (int)(idx / H), h = (int)(idx % H);
    int s = src[e], d = dst[e];
    float v = el[s * H + h] + er[d * H + h];
    v = (v > 0.f) ? v : 0.2f * v;          // leaky_relu, slope 0.2
    eval[idx] = v;
    atomicMax(&emaxu[d * H + h], fmap(v));
}

// ---- edge pass 2: exp(v - max) and segment sum ----------------------------
__global__ void edge_pass2(const int* __restrict__ dst,
                           const unsigned* __restrict__ emaxu,
                           float* __restrict__ eval, float* __restrict__ esum, int H) {
    long long idx = (long long)blockIdx.x * blockDim.x + threadIdx.x;
    if (idx >= (long long)NE * H) return;
    int e = (int)(idx / H), h = (int)(idx % H);
    int d = dst[e];
    float ex = expf(eval[idx] - funmap(emaxu[d * H + h]));
    eval[idx] = ex;
    atomicAdd(&esum[d * H + h], ex);
}

// ---- edge pass 3: alpha = ex / max(den, 1e-9) -----------------------------
__global__ void edge_alpha(const int* __restrict__ dst,
                           float* __restrict__ eval, const float* __restrict__ esum, int H) {
    long long idx = (long long)blockIdx.x * blockDim.x + threadIdx.x;
    if (idx >= (long long)NE * H) return;
    int e = (int)(idx / H), h = (int)(idx % H);
    eval[idx] = eval[idx] / fmaxf(esum[dst[e] * H + h], 1e-9f);
}

// ---- edge pass 4: out[dst,h,:] += alpha * feat[src,h,:] -------------------
// One thread per (edge, head, 4 consecutive channels): b128 gather of the
// L2-resident feat row + 4 coalesced global_atomic_add_f32 scatters.
__global__ void edge_aggregate4(const int* __restrict__ src, const int* __restrict__ dst,
                                const float* __restrict__ feat, int ldf,
                                const float* __restrict__ eval,
                                float* __restrict__ out, int ldo, int H, int Dq /*=D/4*/) {
    long long idx = (long long)blockIdx.x * blockDim.x + threadIdx.x;
    if (idx >= (long long)NE * H * Dq) return;
    int q = (int)(idx % Dq);
    long long t = idx / Dq;
    int h = (int)(t % H);
    int e = (int)(t / H);
    int s = src[e], dn = dst[e];
    float a = eval[(long long)e * H + h];
    const float4 f = *(const float4*)(feat + (size_t)s * ldf + (size_t)h * Dq * 4 + 4 * q);
    float* o = out + (size_t)dn * ldo + (size_t)h * Dq * 4 + 4 * q;
    atomicAdd(o + 0, f.x * a);
    atomicAdd(o + 1, f.y * a);
    atomicAdd(o + 2, f.z * a);
    atomicAdd(o + 3, f.w * a);
}

__global__ void elu_kernel(float* __restrict__ p, long long n) {
    long long i = (long long)blockIdx.x * blockDim.x + threadIdx.x;
    if (i < n) {
        float v = p[i];
        p[i] = (v > 0.f) ? v : expm1f(v);
    }
}

__global__ void copy_out_kernel(const float* __restrict__ in, int ld,
                                float* __restrict__ out, int C, int N) {
    int idx = blockIdx.x * blockDim.x + threadIdx.x;
    if (idx >= N * C) return;
    int n = idx / C, c = idx % C;
    out[idx] = in[(size_t)n * ld + c];
}

// ---------------------------------------------------------------------------
// Host-side orchestration of one GAT conv layer (all enqueues on `stream`).
// ---------------------------------------------------------------------------
struct Scratch {
    float* el; float* er; unsigned* emaxu; float* esum; float* eval;
};

static void gat_layer(const float* in, int lda, int K,
                      const float* W, int ldb, int FoutP,
                      int H, int D,
                      const float* al, const float* ar,
                      float* feat, float* out, int ldo,
                      const int* src, const int* dst,
                      const Scratch& s, bool act, hipStream_t stream) {
    // 1) feat = in @ W  (fp32 WMMA, 16 x 16*NT strip per wave)
    if (FoutP % 64 == 0)
        wmma_gemm_f32<4><<<dim3(NN / 16, FoutP / 64), 32, 0, stream>>>(in, lda, W, ldb, feat, FoutP, K);
    else  // FoutP == 48
        wmma_gemm_f32<3><<<dim3(NN / 16, FoutP / 48), 32, 0, stream>>>(in, lda, W, ldb, feat, FoutP, K);
    // 2) attention logits
    el_er_kernel<<<cdiv((long long)NN * H, 256), 256, 0, stream>>>(feat, FoutP, H, D, al, ar, s.el, s.er, NN);
    // 3) init segment state + zero aggregation target
    init_seg_kernel<<<cdiv((long long)NN * H, 256), 256, 0, stream>>>(s.emaxu, s.esum, NN * H);
    zero_f_kernel<<<cdiv((long long)NN * FoutP, 256), 256, 0, stream>>>(out, (long long)NN * FoutP);
    // 4) edge softmax
    long long EH = (long long)NE * H;
    edge_pass1<<<cdiv(EH, 256), 256, 0, stream>>>(src, dst, s.el, s.er, s.eval, s.emaxu, H);
    edge_pass2<<<cdiv(EH, 256), 256, 0, stream>>>(dst, s.emaxu, s.eval, s.esum, H);
    edge_alpha<<<cdiv(EH, 256), 256, 0, stream>>>(dst, s.eval, s.esum, H);
    // 5) weighted aggregation (vectorized x4)
    edge_aggregate4<<<cdiv(EH * (D / 4), 256), 256, 0, stream>>>(src, dst, feat, FoutP, s.eval, out, ldo, H, D / 4);
    // 6) activation
    if (act)
        elu_kernel<<<cdiv((long long)NN * FoutP, 256), 256, 0, stream>>>(out, (long long)NN * FoutP);
}

extern "C" void kernel_launch(void* const* d_in, const int* in_sizes, int n_in,
                              void* d_out, int out_size, void* d_ws, size_t ws_size,
                              hipStream_t stream) {
    const float* x    = (const float*)d_in[0];
    const int*   src  = (const int*)d_in[1];
    const int*   dst  = (const int*)d_in[2];
    const float* W00  = (const float*)d_in[3];
    const float* a00l = (const float*)d_in[4];
    const float* a00r = (const float*)d_in[5];
    const float* W01  = (const float*)d_in[6];
    const float* a01l = (const float*)d_in[7];
    const float* a01r = (const float*)d_in[8];
    const float* W0f  = (const float*)d_in[9];
    const float* a0fl = (const float*)d_in[10];
    const float* a0fr = (const float*)d_in[11];
    const float* W10  = (const float*)d_in[12];
    const float* a10l = (const float*)d_in[13];
    const float* a10r = (const float*)d_in[14];
    const float* W1f  = (const float*)d_in[15];
    const float* a1fl = (const float*)d_in[16];
    const float* a1fr = (const float*)d_in[17];
    const float* W1o  = (const float*)d_in[18];
    const float* a1ol = (const float*)d_in[19];
    const float* a1or = (const float*)d_in[20];
    float* out = (float*)d_out;

    // ---- workspace carve-up (~170 MB) ----
    char* ws = (char*)d_ws;
    float* FEAT = (float*)ws;           ws += (size_t)NN * 256 * sizeof(float);
    float* A0   = (float*)ws;           ws += (size_t)NN * 256 * sizeof(float);
    float* A1   = (float*)ws;           ws += (size_t)NN * 256 * sizeof(float);
    Scratch s;
    s.el    = (float*)ws;               ws += (size_t)NN * 4 * sizeof(float);
    s.er    = (float*)ws;               ws += (size_t)NN * 4 * sizeof(float);
    s.emaxu = (unsigned*)ws;            ws += (size_t)NN * 4 * sizeof(unsigned);
    s.esum  = (float*)ws;               ws += (size_t)NN * 4 * sizeof(float);
    s.eval  = (float*)ws;               ws += (size_t)NE * 4 * sizeof(float);
    float* Wpad = (float*)ws;           ws += (size_t)H1 * NCP * sizeof(float);

    // ================= branch 0 =================
    // L1: [N,256] -> 4 heads x 64, elu
    gat_layer(x, FIN, FIN, W00, 4 * H1, 4 * H1, 4, H1, a00l, a00r,
              FEAT, A0, 4 * H1, src, dst, s, /*act=*/true, stream);
    // L2: [N,256] -> 1 head x 64, elu
    gat_layer(A0, 4 * H1, 4 * H1, W01, H1, H1, 1, H1, a01l, a01r,
              FEAT, A1, H1, src, dst, s, /*act=*/true, stream);
    // L3: [N,64] -> 1 head x 40 (padded 48), no act; mean over 1 head = id
    pad_w_kernel<<<cdiv(H1 * NCP, 256), 256, 0, stream>>>(W0f, Wpad, H1, NC, NCP);
    gat_layer(A1, H1, H1, Wpad, NCP, NCP, 1, NC, a0fl, a0fr,
              FEAT, A0, NCP, src, dst, s, /*act=*/false, stream);
    copy_out_kernel<<<cdiv((long long)NN * NC, 256), 256, 0, stream>>>(A0, NCP, out, NC, NN);

    // ================= branch 1 =================
    // L1: [N,256] -> 4 heads x 64, elu
    gat_layer(x, FIN, FIN, W10, 4 * H1, 4 * H1, 4, H1, a10l, a10r,
              FEAT, A1, 4 * H1, src, dst, s, /*act=*/true, stream);
    // L2: [N,256] -> 1 head x 64, no act; mean over 1 head = id
    gat_layer(A1, 4 * H1, 4 * H1, W1f, H1, H1, 1, H1, a1fl, a1fr,
              FEAT, A0, H1, src, dst, s, /*act=*/false, stream);
    // L3: [N,64] -> 1 head x 40 (padded 48), elu; [:,0,:] = id
    pad_w_kernel<<<cdiv(H1 * NCP, 256), 256, 0, stream>>>(W1o, Wpad, H1, NC, NCP);
    gat_layer(A0, H1, H1, Wpad, NCP, NCP, 1, NC, a1ol, a1or,
              FEAT, A1, NCP, src, dst, s, /*act=*/true, stream);
    copy_out_kernel<<<cdiv((long long)NN * NC, 256), 256, 0, stream>>>(A1, NCP, out + (size_t)NN * NC, NC, NN);
}